// _DisplacementAgreementLoss_76836964926186
// MI455X (gfx1250) — compile-verified
//
#include <hip/hip_runtime.h>
#include <hip/hip_bf16.h>
#include <math.h>
#include <stdint.h>

typedef float v2f __attribute__((ext_vector_type(2)));
typedef float v8f __attribute__((ext_vector_type(8)));

#define NB   4
#define ND   128
#define NT   8
#define NP   4096
#define NK   10
#define NTM1 7
#define BPP  (NB * NP)   /* 16384 values per frame-pair */

// ---------------------------------------------------------------------------
// Sorted (descending) top-K insertion network, branch-free per slot.
// Keeps earlier (lower) candidate index on exact ties, matching stable top_k.
// ---------------------------------------------------------------------------
__device__ __forceinline__ void topk_insert(float (&vals)[NK], int (&idxs)[NK],
                                            float s, int id) {
#pragma unroll
  for (int j = 0; j < NK; ++j) {
    bool sw = s > vals[j];
    float tv = vals[j];
    int ti = idxs[j];
    vals[j] = sw ? s : tv;
    idxs[j] = sw ? id : ti;
    s  = sw ? tv : s;
    id = sw ? ti : id;
  }
}

// ---------------------------------------------------------------------------
// Kernel 1: per (b, t) — WMMA-based KNN + disp_var + mean cosine similarity.
// grid = (NP/128, NB, NTM1), block = 256 (8 waves of 32; wave owns 16 queries)
//
// Distance scores via V_WMMA_F32_16X16X4_F32:
//   A (16x4) = candidate rows  [x, y, z, xx]      (xx = squared norm)
//   B (4x16) = query   columns [x, y, z, -0.5]
//   D[M][N]  = dot3(c,q) - 0.5*xx_c   == rank-equivalent to neg_dist per column
//
// f32 A-tile layout: lanes 0-15 hold (K0,K1)=(x,y); lanes 16-31 hold (K2,K3).
// C/D layout: lane owns column N=lane%16; VGPR r holds row M=r+8*(lane>=16).
//
// Points are staged into LDS with GLOBAL_LOAD_ASYNC_TO_LDS_B128 (ASYNCcnt),
// then a one-pass fixup writes w := x^2+y^2+z^2.
// ---------------------------------------------------------------------------
__global__ __launch_bounds__(256)
void knn_frame_kernel(const float* __restrict__ enc,
                      const float4* __restrict__ p4,
                      float* __restrict__ disp_var,
                      float* __restrict__ mean_sim) {
  __shared__ float4 spt[NP];  // frame-t points: (x, y, z, xx) — 64 KB

  const int b    = blockIdx.y;
  const int t    = blockIdx.z;
  const int tid  = threadIdx.x;
  const int wave = tid >> 5;
  const int lane = tid & 31;
  const int hl   = lane & 15;
  const bool hi  = lane >= 16;

  const float4* ptsT  = p4 + ((size_t)b * NT + t) * NP;
  const float4* ptsT1 = ptsT + NP;  // frame t+1

  // ---- async copy global -> LDS (16 B per lane per issue, GVS mode) ----
  for (int i = tid; i < NP; i += 256) {
    unsigned lds_addr = (unsigned)(uintptr_t)(&spt[i]);      // LDS byte offset
    unsigned goff     = (unsigned)(i * (int)sizeof(float4)); // global byte off
    asm volatile("global_load_async_to_lds_b128 %0, %1, %2"
                 :
                 : "v"(lds_addr), "v"(goff), "s"(ptsT)
                 : "memory");
  }
  asm volatile("s_wait_asynccnt 0x0" ::: "memory");
  __syncthreads();

  // ---- fixup pass: w := squared norm ----
  for (int i = tid; i < NP; i += 256) {
    float4 v = spt[i];
    spt[i].w = v.x * v.x + v.y * v.y + v.z * v.z;
  }
  __syncthreads();

  const int colBase = (blockIdx.x * 8 + wave) * 16;

  // B-tile (queries), fixed for the whole candidate sweep
  float4 qp = spt[colBase + hl];
  v2f btile;
  btile.x = hi ? qp.z : qp.x;
  btile.y = hi ? -0.5f : qp.y;

  float vals[NK];
  int   idxs[NK];
#pragma unroll
  for (int j = 0; j < NK; ++j) { vals[j] = -3.4e38f; idxs[j] = 0; }

  const int mOff = hi ? 8 : 0;

  // ---- candidate sweep: 4 WMMA tiles per iteration ----
  for (int rowBase = 0; rowBase < NP; rowBase += 64) {
    v2f at[4];
#pragma unroll
    for (int u = 0; u < 4; ++u) {
      float4 cp = spt[rowBase + u * 16 + hl];
      at[u].x = hi ? cp.z : cp.x;
      at[u].y = hi ? cp.w : cp.y;
    }

    v8f acc[4];
#pragma unroll
    for (int u = 0; u < 4; ++u) {
      v8f z = {0.f, 0.f, 0.f, 0.f, 0.f, 0.f, 0.f, 0.f};
      acc[u] = __builtin_amdgcn_wmma_f32_16x16x4_f32(
          /*neg_a=*/false, at[u], /*neg_b=*/false, btile,
          /*c_mod=*/(short)0, z, /*reuse_a=*/false, /*reuse_b=*/false);
    }

#pragma unroll
    for (int u = 0; u < 4; ++u) {
      const int mBase = rowBase + u * 16 + mOff;
#pragma unroll
      for (int m = 0; m < 8; ++m) {
        float s = acc[u][m];
        if (s > vals[NK - 1]) topk_insert(vals, idxs, s, mBase + m);
      }
    }
  }

  // Merge the two half-lane candidate subsets (lane L <-> lane L^16).
  {
    float pv[NK];
    int pi[NK];
#pragma unroll
    for (int j = 0; j < NK; ++j) {
      pv[j] = __shfl_xor(vals[j], 16, 32);
      pi[j] = __shfl_xor(idxs[j], 16, 32);
    }
#pragma unroll
    for (int j = 0; j < NK; ++j) topk_insert(vals, idxs, pv[j], pi[j]);
  }

  // Lanes 0..15: finish their query point q = colBase + lane.
  if (lane < 16) {
    const int q = colBase + lane;

    // ---- displacement variance over the 10 neighbors ----
    float4 qT  = spt[q];
    float4 qT1 = ptsT1[q];
    float dqx = qT1.x - qT.x, dqy = qT1.y - qT.y, dqz = qT1.z - qT.z;
    float dvacc = 0.f;
#pragma unroll
    for (int j = 0; j < NK; ++j) {
      int n = idxs[j];
      float4 nT  = spt[n];
      float4 nT1 = ptsT1[n];
      float ddx = (nT1.x - nT.x) - dqx;
      float ddy = (nT1.y - nT.y) - dqy;
      float ddz = (nT1.z - nT.z) - dqz;
      dvacc += ddx * ddx + ddy * ddy + ddz * ddz;
    }

    // ---- mean cosine similarity over 128-dim features ----
    // enc element (b,d,t,p) at ((b*ND + d)*(NT*NP) + t*NP + p)
    const float* fbase = enc + (size_t)b * ND * (NT * NP) + (size_t)t * NP;
    float qq = 0.f;
    float num[NK], nn[NK];
#pragma unroll
    for (int j = 0; j < NK; ++j) { num[j] = 0.f; nn[j] = 0.f; }

    for (int d = 0; d < ND; ++d) {
      const float* row = fbase + (size_t)d * (NT * NP);
      __builtin_prefetch(row + (size_t)(NT * NP), 0, 1);  // next d-row
      float fq = row[q];
      qq += fq * fq;
#pragma unroll
      for (int j = 0; j < NK; ++j) {
        float fn = row[idxs[j]];
        num[j] += fq * fn;
        nn[j]  += fn * fn;
      }
    }

    float nq = fmaxf(sqrtf(qq), 1e-8f);
    float cs = 0.f;
#pragma unroll
    for (int j = 0; j < NK; ++j)
      cs += num[j] / (nq * fmaxf(sqrtf(nn[j]), 1e-8f));

    size_t o = ((size_t)t * NB + b) * NP + q;
    disp_var[o] = dvacc * (1.0f / NK);
    mean_sim[o] = cs * (1.0f / NK);
  }
}

// ---------------------------------------------------------------------------
// Kernel 2: per frame t — exact median of 16384 disp_var values via a
// bitonic sort in 64 KB LDS. scale[t] = max(sorted[(N-1)/2], 1e-6).
// ---------------------------------------------------------------------------
__global__ __launch_bounds__(1024)
void median_kernel(const float* __restrict__ disp_var, float* __restrict__ scale) {
  __shared__ float s[BPP];
  const int t = blockIdx.x;
  const int tid = threadIdx.x;
  const float* src = disp_var + (size_t)t * BPP;

  for (int i = tid; i < BPP; i += 1024) s[i] = src[i];
  __syncthreads();

  for (int k = 2; k <= BPP; k <<= 1) {
    for (int j = k >> 1; j > 0; j >>= 1) {
      for (int i = tid; i < BPP; i += 1024) {
        int ixj = i ^ j;
        if (ixj > i) {
          bool up = ((i & k) == 0);
          float a = s[i], c = s[ixj];
          if ((a > c) == up) { s[i] = c; s[ixj] = a; }
        }
      }
      __syncthreads();
    }
  }

  if (tid == 0) scale[t] = fmaxf(s[(BPP - 1) / 2], 1e-6f);
}

// ---------------------------------------------------------------------------
// Kernel 3: per frame t — mean over B*P of exp(-dv/scale)*(1 - mean_sim).
// ---------------------------------------------------------------------------
__global__ __launch_bounds__(256)
void frame_reduce_kernel(const float* __restrict__ disp_var,
                         const float* __restrict__ mean_sim,
                         const float* __restrict__ scale,
                         float* __restrict__ partial) {
  __shared__ float red[256];
  const int t = blockIdx.x;
  const int tid = threadIdx.x;
  const float sc = scale[t];
  const float* dv = disp_var + (size_t)t * BPP;
  const float* ms = mean_sim + (size_t)t * BPP;

  float acc = 0.f;
  for (int i = tid; i < BPP; i += 256)
    acc += expf(-dv[i] / sc) * (1.0f - ms[i]);

  red[tid] = acc;
  __syncthreads();
  for (int w = 128; w > 0; w >>= 1) {
    if (tid < w) red[tid] += red[tid + w];
    __syncthreads();
  }
  if (tid == 0) partial[t] = red[0] * (1.0f / BPP);
}

// ---------------------------------------------------------------------------
// Kernel 4: combine the 7 per-frame means in a fixed order (deterministic).
// ---------------------------------------------------------------------------
__global__ void final_kernel(const float* __restrict__ partial,
                             float* __restrict__ out) {
  if (threadIdx.x == 0 && blockIdx.x == 0) {
    float s = 0.f;
    for (int t = 0; t < NTM1; ++t) s += partial[t];
    out[0] = s / (float)NTM1;
  }
}

// ---------------------------------------------------------------------------
extern "C" void kernel_launch(void* const* d_in, const int* in_sizes, int n_in,
                              void* d_out, int out_size, void* d_ws, size_t ws_size,
                              hipStream_t stream) {
  const float*  enc = (const float*)d_in[0];   // [B, D, T*P] f32
  const float4* p4  = (const float4*)d_in[1];  // [B, T, P, 4] f32
  (void)in_sizes; (void)n_in; (void)out_size; (void)ws_size;

  float* ws = (float*)d_ws;
  float* dv = ws;                                // NTM1*BPP floats
  float* ms = dv + (size_t)NTM1 * BPP;           // NTM1*BPP floats
  float* sc = ms + (size_t)NTM1 * BPP;           // NTM1 floats
  float* pt = sc + NTM1;                         // NTM1 floats
  float* out = (float*)d_out;

  dim3 g1(NP / 128, NB, NTM1);
  knn_frame_kernel<<<g1, 256, 0, stream>>>(enc, p4, dv, ms);
  median_kernel<<<NTM1, 1024, 0, stream>>>(dv, sc);
  frame_reduce_kernel<<<NTM1, 256, 0, stream>>>(dv, ms, sc, pt);
  final_kernel<<<1, 1, 0, stream>>>(pt, out);
}